// GCN_34643206209888
// MI455X (gfx1250) — compile-verified
//
#include <hip/hip_runtime.h>

#define NN   100000
#define EE   1600000
#define CINC 16
#define HC   128
#define COC  10
#define GC   16
#define EPSF 1e-5f

typedef float v2f __attribute__((ext_vector_type(2)));
typedef float v8f __attribute__((ext_vector_type(8)));

// ---------------------------------------------------------------- utilities
__global__ void fill_k(float* __restrict__ p, float v, long n) {
  long i = blockIdx.x * (long)blockDim.x + threadIdx.x;
  if (i < n) p[i] = v;
}

__global__ void deg_k(const int* __restrict__ dst, float* __restrict__ deg, int E) {
  int i = blockIdx.x * blockDim.x + threadIdx.x;
  if (i < E) atomicAdd(&deg[dst[i]], 1.0f);
}

__global__ void rsqrt_k(float* __restrict__ d, int n) {
  int i = blockIdx.x * blockDim.x + threadIdx.x;
  if (i < n) d[i] = rsqrtf(d[i]);
}

// -------------------------------------------------- fused bias+ReLU+BN-stats
// Each thread owns channel c = tid % C, strides over rows. Block-level LDS
// reduction -> one atomic per channel per block (no hot-spot on 128 floats).
__global__ void stat_k(float* __restrict__ h, const float* __restrict__ bias,
                       float* __restrict__ sums, float* __restrict__ sumsq,
                       int N, int C, int fuse) {
  __shared__ float ls[256];
  __shared__ float lss[256];
  int tid = threadIdx.x;
  int c = tid % C;
  int rpb = 256 / C;
  int r = blockIdx.x * rpb + tid / C;
  int stride = gridDim.x * rpb;
  float b = fuse ? bias[c] : 0.0f;
  float s = 0.0f, ss = 0.0f;
  for (; r < N; r += stride) {
    float v = h[(long)r * C + c];
    if (fuse) { v = fmaxf(v + b, 0.0f); h[(long)r * C + c] = v; }
    s += v; ss += v * v;
  }
  ls[tid] = s; lss[tid] = ss;
  __syncthreads();
  if (tid < C) {
    for (int j = 1; j < rpb; ++j) { s += ls[tid + j * C]; ss += lss[tid + j * C]; }
    atomicAdd(&sums[tid], s);
    atomicAdd(&sumsq[tid], ss);
  }
}

__global__ void bnnorm_k(const float* __restrict__ h, float* __restrict__ out,
                         const float* __restrict__ sums, const float* __restrict__ sumsq,
                         const float* __restrict__ gamma, const float* __restrict__ beta,
                         long total, int C, float invN) {
  long i = blockIdx.x * (long)blockDim.x + threadIdx.x;
  if (i >= total) return;
  int c = (int)(i % C);
  float m = sums[c] * invN;
  float var = sumsq[c] * invN - m * m;
  out[i] = gamma[c] * (h[i] - m) * rsqrtf(var + EPSF) + beta[c];
}

// ------------------------------------------------------------ WMMA fp32 GEMM
// B = A[N,K] x W[K,128].  One wave computes a 16(M) x 128(N) strip using
// 8 accumulator tiles of V_WMMA_F32_16X16X4_F32, A fragment reused across the
// 8 N-tiles, W staged in LDS (64 KB of the 320 KB/WGP).
__global__ void gemm_wmma_k(const float* __restrict__ A, const float* __restrict__ W,
                            float* __restrict__ B, int N, int K) {
  __shared__ float Wl[HC * HC];            // only K*128 entries used
  for (int i = threadIdx.x; i < K * HC; i += blockDim.x) Wl[i] = W[i];
  __syncthreads();

  int lane = threadIdx.x & 31;
  int wave = threadIdx.x >> 5;
  int tile = blockIdx.x * 8 + wave;        // 256 threads = 8 waves/block
  int numTiles = N >> 4;                   // N divisible by 16
  if (tile >= numTiles) return;            // wave-uniform: EXEC all-ones at WMMA

  int nl = lane & 15;
  int hi = lane >> 4;                      // ISA A-layout: lanes 16-31 hold K+2
  const float* arow = A + (long)(tile * 16 + nl) * K;

  v8f acc0 = {0,0,0,0,0,0,0,0}, acc1 = acc0, acc2 = acc0, acc3 = acc0;
  v8f acc4 = acc0, acc5 = acc0, acc6 = acc0, acc7 = acc0;
  v8f acc[8] = {acc0, acc1, acc2, acc3, acc4, acc5, acc6, acc7};

  for (int kk = 0; kk < K; kk += 4) {
    v2f a;
    a.x = arow[kk + 2 * hi];
    a.y = arow[kk + 2 * hi + 1];
#pragma unroll
    for (int t = 0; t < 8; ++t) {
      v2f b;
      b.x = Wl[(kk + 2 * hi) * HC + t * 16 + nl];
      b.y = Wl[(kk + 2 * hi + 1) * HC + t * 16 + nl];
      acc[t] = __builtin_amdgcn_wmma_f32_16x16x4_f32(
          false, a, false, b, (short)0, acc[t], false, false);
    }
  }
  // C/D layout: VGPR j -> row (j + 8*hi), col = nl within tile
#pragma unroll
  for (int t = 0; t < 8; ++t) {
#pragma unroll
    for (int j = 0; j < 8; ++j) {
      B[(long)(tile * 16 + j + 8 * hi) * HC + t * 16 + nl] = acc[t][j];
    }
  }
}

// ------------------------------------------------- edge gather/scale/scatter
// One wave per edge (plus N virtual self-loop edges); each lane handles 4
// channels: float4 gather of t[src], 4x global_atomic_add_f32 into acc[dst].
__global__ void scatter_k(const float* __restrict__ t, const float* __restrict__ inv,
                          const int* __restrict__ src, const int* __restrict__ dst,
                          float* __restrict__ acc, int E, int N) {
  long long gid = blockIdx.x * (long long)blockDim.x + threadIdx.x;
  int lane = (int)(gid & 31);
  long long e = gid >> 5;
  if (e >= (long long)E + N) return;
  int s, d; float w;
  if (e < E) {
    s = src[e]; d = dst[e];
    w = inv[s] * inv[d];
  } else {
    s = d = (int)(e - E);
    float iv = inv[s];
    w = iv * iv;                            // self-loop weight
  }
  const float4 v = *(const float4*)(t + (long)s * HC + lane * 4);
  float* o = acc + (long)d * HC + lane * 4;
  atomicAdd(o + 0, v.x * w);
  atomicAdd(o + 1, v.y * w);
  atomicAdd(o + 2, v.z * w);
  atomicAdd(o + 3, v.w * w);
}

// --------------------------------------------------------------- mean pool
__global__ void cnt_k(const int* __restrict__ batch, float* __restrict__ cnt, int N) {
  __shared__ float lc[GC];
  if (threadIdx.x < GC) lc[threadIdx.x] = 0.0f;
  __syncthreads();
  int i = blockIdx.x * blockDim.x + threadIdx.x;
  if (i < N) atomicAdd(&lc[batch[i]], 1.0f);
  __syncthreads();
  if (threadIdx.x < GC) atomicAdd(&cnt[threadIdx.x], lc[threadIdx.x]);
}

__global__ void pool_k(const float* __restrict__ h, const int* __restrict__ batch,
                       float* __restrict__ pooled, int N) {
  long long gid = blockIdx.x * (long long)blockDim.x + threadIdx.x;
  int lane = (int)(gid & 31);
  long long node = gid >> 5;
  if (node >= N) return;
  int g = batch[node];
  const float4 v = *(const float4*)(h + (long)node * HC + lane * 4);
  float* o = pooled + (long)g * HC + lane * 4;
  atomicAdd(o + 0, v.x);
  atomicAdd(o + 1, v.y);
  atomicAdd(o + 2, v.z);
  atomicAdd(o + 3, v.w);
}

__global__ void pooldiv_k(float* __restrict__ pooled, const float* __restrict__ cnt) {
  int i = blockIdx.x * blockDim.x + threadIdx.x;
  if (i < GC * HC) pooled[i] /= fmaxf(cnt[i / HC], 1.0f);
}

// ----------------------------------------- fused FC5 + ReLU + BN5 + FC6 tail
__global__ void final_k(const float* __restrict__ pooled,
                        const float* __restrict__ W5, const float* __restrict__ b5,
                        const float* __restrict__ g5, const float* __restrict__ be5,
                        const float* __restrict__ W6, const float* __restrict__ b6,
                        float* __restrict__ out) {
  __shared__ float h5[GC * HC];
  int tid = threadIdx.x;
  if (tid < HC) {
    float vals[GC];
    float mean = 0.0f;
    for (int g = 0; g < GC; ++g) {
      float s = b5[tid];
      for (int k = 0; k < HC; ++k) s += pooled[g * HC + k] * W5[k * HC + tid];
      s = fmaxf(s, 0.0f);
      vals[g] = s;
      mean += s;
    }
    mean *= (1.0f / GC);
    float var = 0.0f;
    for (int g = 0; g < GC; ++g) { float d = vals[g] - mean; var += d * d; }
    var *= (1.0f / GC);
    float is = rsqrtf(var + EPSF);
    for (int g = 0; g < GC; ++g)
      h5[g * HC + tid] = g5[tid] * (vals[g] - mean) * is + be5[tid];
  }
  __syncthreads();
  if (tid < GC * COC) {
    int g = tid / COC, o = tid % COC;
    float s = b6[o];
    for (int k = 0; k < HC; ++k) s += h5[g * HC + k] * W6[k * COC + o];
    out[g * COC + o] = s;
  }
}

// ---------------------------------------------------------------- launcher
extern "C" void kernel_launch(void* const* d_in, const int* in_sizes, int n_in,
                              void* d_out, int out_size, void* d_ws, size_t ws_size,
                              hipStream_t stream) {
  const float* x     = (const float*)d_in[0];
  const int*   ei    = (const int*)d_in[1];
  const int*   batch = (const int*)d_in[2];
  const float* Wc[4] = {(const float*)d_in[3], (const float*)d_in[5],
                        (const float*)d_in[7], (const float*)d_in[9]};
  const float* bc[4] = {(const float*)d_in[4], (const float*)d_in[6],
                        (const float*)d_in[8], (const float*)d_in[10]};
  const float* W5 = (const float*)d_in[11];
  const float* b5 = (const float*)d_in[12];
  const float* W6 = (const float*)d_in[13];
  const float* b6 = (const float*)d_in[14];
  const float* gamma0 = (const float*)d_in[15];
  const float* beta0  = (const float*)d_in[16];
  const float* gamma5 = (const float*)d_in[17];
  const float* beta5  = (const float*)d_in[18];
  const float* gc[4] = {(const float*)d_in[19], (const float*)d_in[21],
                        (const float*)d_in[23], (const float*)d_in[25]};
  const float* bec[4] = {(const float*)d_in[20], (const float*)d_in[22],
                         (const float*)d_in[24], (const float*)d_in[26]};
  const int* src = ei;
  const int* dst = ei + EE;
  float* out = (float*)d_out;

  // workspace layout (floats)
  float* ws     = (float*)d_ws;
  float* hn     = ws;                         // [N,128] normalized activations
  float* tbuf   = hn   + (long)NN * HC;       // [N,128] GEMM output
  float* accb   = tbuf + (long)NN * HC;       // [N,128] scatter accumulator
  float* xn     = accb + (long)NN * HC;       // [N,16]  BN0(x)
  float* inv    = xn   + (long)NN * CINC;     // [N] deg -> rsqrt(deg)
  float* sums   = inv  + NN;                  // [128]
  float* sumsq  = sums + HC;                  // [128]
  float* pooled = sumsq + HC;                 // [16,128]
  float* cnt    = pooled + GC * HC;           // [16]

  const int T = 256;
  const long NH = (long)NN * HC;
  const int gNH   = (int)((NH + T - 1) / T);
  const int gN16  = (int)(((long)NN * CINC + T - 1) / T);
  const int gN    = (NN + T - 1) / T;
  const int gE    = (EE + T - 1) / T;
  const int gScat = (int)(((long long)(EE + NN) * 32 + T - 1) / T);
  const int gPool = (int)(((long long)NN * 32 + T - 1) / T);
  const int gGemm = ((NN >> 4) + 7) / 8;      // 8 waves/block, 16 rows/wave

  // degrees (init 1.0 == self-loop), then inv = rsqrt(deg)
  fill_k<<<gN, T, 0, stream>>>(inv, 1.0f, NN);
  deg_k<<<gE, T, 0, stream>>>(dst, inv, EE);
  rsqrt_k<<<gN, T, 0, stream>>>(inv, NN);

  // BN0 on x -> xn
  fill_k<<<1, T, 0, stream>>>(sums, 0.0f, 2 * HC);
  stat_k<<<512, T, 0, stream>>>((float*)x, nullptr, sums, sumsq, NN, CINC, 0);
  bnnorm_k<<<gN16, T, 0, stream>>>(x, xn, sums, sumsq, gamma0, beta0,
                                   (long)NN * CINC, CINC, 1.0f / NN);

  // 4 GCN conv layers
  for (int l = 0; l < 4; ++l) {
    const float* in = (l == 0) ? xn : hn;
    const int K = (l == 0) ? CINC : HC;
    gemm_wmma_k<<<gGemm, T, 0, stream>>>(in, Wc[l], tbuf, NN, K);
    fill_k<<<gNH, T, 0, stream>>>(accb, 0.0f, NH);
    scatter_k<<<gScat, T, 0, stream>>>(tbuf, inv, src, dst, accb, EE, NN);
    fill_k<<<1, T, 0, stream>>>(sums, 0.0f, 2 * HC);
    stat_k<<<512, T, 0, stream>>>(accb, bc[l], sums, sumsq, NN, HC, 1);
    bnnorm_k<<<gNH, T, 0, stream>>>(accb, hn, sums, sumsq, gc[l], bec[l],
                                    NH, HC, 1.0f / NN);
  }

  // global mean pool
  fill_k<<<1, T, 0, stream>>>(pooled, 0.0f, GC * HC + GC);
  cnt_k<<<gN, T, 0, stream>>>(batch, cnt, NN);
  pool_k<<<gPool, T, 0, stream>>>(hn, batch, pooled, NN);
  pooldiv_k<<<(GC * HC + T - 1) / T, T, 0, stream>>>(pooled, cnt);

  // FC5 + ReLU + BN5 + FC6
  final_k<<<1, T, 0, stream>>>(pooled, W5, b5, gamma5, beta5, W6, b6, out);
}